// LanczosUpsampling_55843164783389
// MI455X (gfx1250) — compile-verified
//
#include <hip/hip_runtime.h>

// Lanczos 8x upsampling (a=4, 9 taps), fused separable passes as two
// V_WMMA_F32_16X16X4_F32 matmuls per 16x16 output tile.
//
//   out[i,j] = sum_q wv(q, i&7) * H[reflectV(i+q)>>3, j]
//   H[yi,j]  = sum_p wh(p, j&7) * img[yi, reflectH(j+p)>>3]
//
// Each wave owns a strip of 16 tiles at fixed i0:
//  - 9 raw tap weights wq[] computed once per lane (phase == m&7 everywhere)
//  - vertical combined weights (A operand) once per wave
//  - interior horizontal combined weights (B operand) once per wave;
//    only tx==0 / tx==127 recompute with reflection (wave-uniform branch)
// Per tile: 2 loads -> WMMA (vertical) -> 16x4 LDS transpose -> WMMA
// (horizontal) -> 8 clause stores. Store-bandwidth bound (~128 MB out).
//
// The transpose buffer is PER-WAVE private and CDNA5 LDS ops from one wave
// are processed in order (single in-order DScnt stream), so no block barrier
// is needed between the ds_store and ds_load — only a compiler reordering
// fence. This removes per-tile s_barrier_* and decouples the 8 waves.

typedef float v2f __attribute__((ext_vector_type(2)));
typedef float v8f __attribute__((ext_vector_type(8)));

#define HI 256
#define WI 256
#define HO 2048
#define WO 2048
#define NPLANE 8           // B*C = 4*2
#define TILES_X 128
#define TILES_Y 128
#define CHUNK 16           // tiles per wave (one strip)
#define NCHUNK (TILES_X / CHUNK)   // 8

// reflect-pad (pad=4, jnp 'reflect') index in upsampled coords -> orig index
__device__ __forceinline__ int reflect_up(int q, int n_up) {
    if (q < 4) return 4 - q;
    if (q >= n_up + 4) return 2 * n_up + 2 - q;
    return q - 4;
}

__global__ __launch_bounds__(256) void lanczos_up8_wmma(
    const float* __restrict__ img, float* __restrict__ out) {
    __shared__ float lds[8][16 * 4];   // per-wave 16x4 transpose buffer

    const int lane = threadIdx.x & 31;
    const int wave = threadIdx.x >> 5;
    const int wid  = blockIdx.x * 8 + wave;   // 0 .. 8191
    const int plane = wid >> 10;              // / (TILES_Y * NCHUNK)
    const int rem   = wid & 1023;
    const int ty    = rem >> 3;               // 0..127 tile row
    const int tx0   = (rem & 7) * CHUNK;      // strip base tile col
    const int i0    = ty << 4;

    const int m  = lane & 15;                 // M / N index
    const int hi = lane >> 4;                 // lane half
    const int k0 = hi << 1;                   // K indices held by this lane
    const int k1 = k0 + 1;                    // (f32 A/B layout: v + 2*(lane/16))

    // ---- 9 raw Lanczos tap weights for this lane's phase (phase == m&7) ----
    const int ph = m & 7;
    float wq[9];
    #pragma unroll
    for (int q = 0; q < 9; ++q) {
        float t = (1e-8f + 0.125f * (float)ph) + (float)(q - 4);
        float pt  = 3.14159265358979f * t;
        float pt4 = pt * 0.25f;
        wq[q] = (__sinf(pt) / pt) * (__sinf(pt4) / pt4);
    }

    // ---- vertical combined weights (general: handles top/bottom reflection) ----
    const int yLo = max(0, (i0 - 4) >> 3);
    float wa0 = 0.f, wa1 = 0.f;
    #pragma unroll
    for (int q = 0; q < 9; ++q) {
        int r = (reflect_up(i0 + m + q, HO) >> 3) - yLo;
        if (r == k0) wa0 += wq[q];
        if (r == k1) wa1 += wq[q];
    }
    v2f a1; a1.x = wa0; a1.y = wa1;

    // ---- interior horizontal combined weights (closed form, tile-invariant) ----
    float wi0 = 0.f, wi1 = 0.f;
    #pragma unroll
    for (int q = 0; q < 9; ++q) {
        int r = ((m + q - 4) >> 3) + 1;       // in 0..3
        if (r == k0) wi0 += wq[q];
        if (r == k1) wi1 += wq[q];
    }

    // ---- wave-invariant bases ----
    const float* ipl = img + (size_t)plane * (HI * WI);
    const float* rp0 = ipl + min(HI - 1, yLo + k0) * WI;   // this lane's K rows
    const float* rp1 = ipl + min(HI - 1, yLo + k1) * WI;
    float* sl = lds[wave];
    float* op = out + (size_t)plane * ((size_t)HO * WO) + (size_t)i0 * WO + m;

    for (int t = 0; t < CHUNK; ++t) {
        const int tx = tx0 + t;
        const int j0 = tx << 4;
        const int xLo = max(0, (j0 - 4) >> 3);
        const int colg = min(WI - 1, xLo + m);   // clamped cols carry zero weight

        // ===== pass 1 (vertical): D1[m][c] = sum_k WV[m][k] * img[yLo+k][xLo+c]
        v2f b1; b1.x = rp0[colg]; b1.y = rp1[colg];
        v8f accV = {};
        accV = __builtin_amdgcn_wmma_f32_16x16x4_f32(
            false, a1, false, b1, (short)0, accV, false, false);

        // D-layout -> A-layout transpose; only input-cols 0..3 are pass-2 K.
        // Same-wave LDS ops are in-order on CDNA5; buffer is wave-private, so
        // no block barrier needed — just stop the compiler from reordering.
        if (m < 4) {
            #pragma unroll
            for (int d = 0; d < 8; ++d)
                sl[(d + 8 * hi) * 4 + m] = accV[d];
        }
        asm volatile("" ::: "memory");

        v2f a2; a2.x = sl[m * 4 + k0]; a2.y = sl[m * 4 + k1];

        v2f b2; b2.x = wi0; b2.y = wi1;
        if (tx == 0 || tx == TILES_X - 1) {      // wave-uniform edge fixup
            float e0 = 0.f, e1 = 0.f;
            #pragma unroll
            for (int q = 0; q < 9; ++q) {
                int c = (reflect_up(j0 + m + q, WO) >> 3) - xLo;
                if (c == k0) e0 += wq[q];
                if (c == k1) e1 += wq[q];
            }
            b2.x = e0; b2.y = e1;
        }

        // ===== pass 2 (horizontal): D2[m][n] = sum_c V[m][c] * WH[c][n]
        v8f accO = {};
        accO = __builtin_amdgcn_wmma_f32_16x16x4_f32(
            false, a2, false, b2, (short)0, accO, false, false);

        #pragma unroll
        for (int d = 0; d < 8; ++d)
            op[(size_t)((d + 8 * hi) * WO) + j0] = accO[d];

        asm volatile("" ::: "memory");   // keep next tile's LDS writes after this read
    }
}

extern "C" void kernel_launch(void* const* d_in, const int* in_sizes, int n_in,
                              void* d_out, int out_size, void* d_ws, size_t ws_size,
                              hipStream_t stream) {
    (void)in_sizes; (void)n_in; (void)d_ws; (void)ws_size; (void)out_size;
    const float* img = (const float*)d_in[0];   // (4,2,256,256) f32
    float* out = (float*)d_out;                 // (4,2,2048,2048) f32
    // 8 planes * 128 tile rows * 8 strips, 8 waves (strips) per block
    const int nblocks = (NPLANE * TILES_Y * NCHUNK) / 8;   // 1024
    lanczos_up8_wmma<<<nblocks, 256, 0, stream>>>(img, out);
}